// MultiHeadAttentionWeightedAverage_6408091206155
// MI455X (gfx1250) — compile-verified
//
#include <hip/hip_runtime.h>
#include <math.h>

// ---------------------------------------------------------------------------
// MultiHeadAttentionWeightedAverage, algebraically reduced:
//   qk[k][h]   = (1/sqrt(64)) * sum_j Wk[h*64+j][k] * query[h][j]
//   score[r,h] = sum_k x[r][k] * qk[k][h]              (r = b*4096+n)   [WMMA]
//   A[b,h,:]   = softmax_n(score)
//   s[b,h,:]   = sum_n A[b,h,n] * x[b*4096+n][:]                        [WMMA]
//   pooled[b,i]= dot(Wv[i,:], s[b, i/64, :])
//   out[b,i]   = dot(Wout[i,:], pooled[b,:]) + bout[i]
// ---------------------------------------------------------------------------

typedef __attribute__((ext_vector_type(2))) float v2f;
typedef __attribute__((ext_vector_type(8))) float v8f;

#define D 1024
#define H 16
#define N 4096
#define B 16
#define NROWS (B * N)            // 65536
#define INV_SCALE 0.125f         // 1/sqrt(64)

// --- K0: fold query into Wk:  qkmat[k*16 + h]  (k-major so WMMA B-loads coalesce)
__global__ void qk_kernel(const float* __restrict__ Wk, const float* __restrict__ query,
                          float* __restrict__ qkmat) {
    int t = blockIdx.x * blockDim.x + threadIdx.x;   // 16384 threads
    int h = t >> 10;
    int k = t & 1023;
    const float* wcol = Wk + (long)h * 64 * D + k;   // Wk[h*64 + j][k], stride D
    const float* q = query + h * 64;
    float acc = 0.f;
    #pragma unroll 8
    for (int j = 0; j < 64; ++j) acc += wcol[(long)j * D] * q[j];
    qkmat[k * H + h] = acc * INV_SCALE;
}

// --- K1: scores = X(65536x1024) @ qkmat(1024x16), one wave per 16-row tile.
// V_WMMA_F32_16X16X4_F32 layouts (wave32):
//   A 16x4 : lane m=lane&15 -> row M=m ; lane>>4 selects K-pair; v2f = {K, K+1}
//   B 4x16 : VGPR0 = (K = lane>>4, N = lane&15), VGPR1 = (K+2, same N)
//   C 16x16: VGPR r -> M = r + 8*(lane>>4), N = lane&15
__global__ void score_kernel(const float* __restrict__ x, const float* __restrict__ qkmat,
                             float* __restrict__ scores) {
    const int lane = threadIdx.x & 31;
    const int tile = blockIdx.x * (blockDim.x >> 5) + (threadIdx.x >> 5); // 4096 tiles
    const long rowBase = (long)tile * 16;
    const int m  = lane & 15;
    const int kh = lane >> 4;
    const float* xrow = x + (rowBase + m) * D + kh * 2;
    v8f c = {0.f, 0.f, 0.f, 0.f, 0.f, 0.f, 0.f, 0.f};
    for (int k = 0; k < D; k += 4) {
        v2f a = *(const v2f*)(xrow + k);               // x[row][k + kh*2 .. +1]
        v2f b;
        b.x = qkmat[(k + kh) * H + m];                 // (K=k+kh,   N=m)
        b.y = qkmat[(k + 2 + kh) * H + m];             // (K=k+2+kh, N=m)
        c = __builtin_amdgcn_wmma_f32_16x16x4_f32(false, a, false, b,
                                                  (short)0, c, false, false);
    }
    #pragma unroll
    for (int r = 0; r < 8; ++r) {
        int row = r + kh * 8;
        scores[(rowBase + row) * H + m] = c[r];
    }
}

// --- K2: softmax over n for each (b,h); writes probs[(b*16+h)*4096 + n]
__global__ void softmax_kernel(const float* __restrict__ scores, float* __restrict__ probs) {
    const int b = blockIdx.x >> 4;
    const int h = blockIdx.x & 15;
    const float* sp = scores + (long)b * N * H + h;    // stride H over n
    float* pp = probs + (long)blockIdx.x * N;
    __shared__ float sred[8];
    __shared__ float bcast;
    const int tid = threadIdx.x, lane = tid & 31, wid = tid >> 5;

    float mx = -1e30f;
    for (int n = tid; n < N; n += 256) mx = fmaxf(mx, sp[(long)n * H]);
    for (int off = 16; off; off >>= 1) mx = fmaxf(mx, __shfl_xor(mx, off, 32));
    if (lane == 0) sred[wid] = mx;
    __syncthreads();
    if (tid == 0) {
        float v = sred[0];
        for (int i = 1; i < 8; ++i) v = fmaxf(v, sred[i]);
        bcast = v;
    }
    __syncthreads();
    const float gmax = bcast;

    float sum = 0.f;
    for (int n = tid; n < N; n += 256) {
        float e = __expf(sp[(long)n * H] - gmax);
        pp[n] = e;
        sum += e;
    }
    for (int off = 16; off; off >>= 1) sum += __shfl_xor(sum, off, 32);
    __syncthreads();                 // sred reuse: max-phase reads complete
    if (lane == 0) sred[wid] = sum;
    __syncthreads();
    if (tid == 0) {
        float v = 0.f;
        for (int i = 0; i < 8; ++i) v += sred[i];
        bcast = 1.0f / v;
    }
    __syncthreads();
    const float inv = bcast;
    for (int n = tid; n < N; n += 256) pp[n] *= inv;
}

// --- K3: s_b(16x1024) = A_b(16heads x 4096) @ X_b(4096x1024); wave per 16-col tile
__global__ void wsum_kernel(const float* __restrict__ x, const float* __restrict__ probs,
                            float* __restrict__ svec) {
    const int lane = threadIdx.x & 31;
    const int gw = blockIdx.x * (blockDim.x >> 5) + (threadIdx.x >> 5);  // 0..1023
    const int b  = gw >> 6;           // 16 batches
    const int ct = gw & 63;           // 64 column tiles of 16
    const int col0 = ct * 16;
    const int m  = lane & 15;
    const int kh = lane >> 4;
    const float* pb = probs + (long)b * H * N;        // [h][n]
    const float* xb = x + (long)b * N * D;
    v8f c = {0.f, 0.f, 0.f, 0.f, 0.f, 0.f, 0.f, 0.f};
    for (int n = 0; n < N; n += 4) {
        v2f a;                                        // A: M=head=m, K=n+kh*2..+1
        a.x = pb[(long)m * N + n + kh * 2];
        a.y = pb[(long)m * N + n + kh * 2 + 1];
        v2f bb;                                       // B: (K=n+kh, n+2+kh; N=col0+m)
        bb.x = xb[(long)(n + kh) * D + col0 + m];
        bb.y = xb[(long)(n + 2 + kh) * D + col0 + m];
        c = __builtin_amdgcn_wmma_f32_16x16x4_f32(false, a, false, bb,
                                                  (short)0, c, false, false);
    }
    #pragma unroll
    for (int r = 0; r < 8; ++r) {
        int head = r + kh * 8;
        svec[((long)b * H + head) * D + col0 + m] = c[r];
    }
}

// --- K4: pooled[b][i] = dot(Wv[i,:], s[b, i/64, :]); one wave per output
__global__ void pooled_kernel(const float* __restrict__ Wv, const float* __restrict__ svec,
                              float* __restrict__ pooled) {
    const int gw = blockIdx.x * (blockDim.x >> 5) + (threadIdx.x >> 5);  // 16384 waves
    const int b = gw >> 10;
    const int idx = gw & 1023;
    const int h = idx >> 6;
    const float* w = Wv + (long)idx * D;
    const float* s = svec + ((long)b * H + h) * D;
    const int lane = threadIdx.x & 31;
    float acc = 0.f;
    for (int d = lane; d < D; d += 32) acc += w[d] * s[d];
    for (int off = 16; off; off >>= 1) acc += __shfl_xor(acc, off, 32);
    if (lane == 0) pooled[(long)b * D + idx] = acc;
}

// --- K5: out[b][i] = dot(Wout[i,:], pooled[b,:]) + bout[i]; one wave per output
__global__ void out_kernel(const float* __restrict__ Wout, const float* __restrict__ bout,
                           const float* __restrict__ pooled, float* __restrict__ out) {
    const int gw = blockIdx.x * (blockDim.x >> 5) + (threadIdx.x >> 5);  // 16384 waves
    const int b = gw >> 10;
    const int idx = gw & 1023;
    const float* w = Wout + (long)idx * D;
    const float* p = pooled + (long)b * D;
    const int lane = threadIdx.x & 31;
    float acc = 0.f;
    for (int d = lane; d < D; d += 32) acc += w[d] * p[d];
    for (int off = 16; off; off >>= 1) acc += __shfl_xor(acc, off, 32);
    if (lane == 0) out[(long)b * D + idx] = acc + bout[idx];
}

extern "C" void kernel_launch(void* const* d_in, const int* in_sizes, int n_in,
                              void* d_out, int out_size, void* d_ws, size_t ws_size,
                              hipStream_t stream) {
    const float* x     = (const float*)d_in[0];
    const float* Wk    = (const float*)d_in[1];
    const float* Wv    = (const float*)d_in[2];
    const float* query = (const float*)d_in[3];
    const float* Wout  = (const float*)d_in[4];
    const float* bout  = (const float*)d_in[5];
    float* out = (float*)d_out;

    float* ws     = (float*)d_ws;
    float* qkmat  = ws;                    // 1024*16        = 16384
    float* scores = qkmat + 16384;         // 65536*16       = 1048576
    float* probs  = scores + 1048576;      // 256*4096       = 1048576
    float* svec   = probs + 1048576;       // 16*16*1024     = 262144
    float* pooled = svec + 262144;         // 16*1024        = 16384
    (void)ws_size; (void)in_sizes; (void)n_in; (void)out_size;

    qk_kernel     <<<  64, 256, 0, stream>>>(Wk, query, qkmat);
    score_kernel  <<<1024, 128, 0, stream>>>(x, qkmat, scores);   // 4096 waves
    softmax_kernel<<< 256, 256, 0, stream>>>(scores, probs);
    wsum_kernel   <<< 256, 128, 0, stream>>>(x, probs, svec);     // 1024 waves
    pooled_kernel <<<2048, 256, 0, stream>>>(Wv, svec, pooled);
    out_kernel    <<<2048, 256, 0, stream>>>(Wout, bout, pooled, out);
}